// SparseAttention_8478265442446
// MI455X (gfx1250) — compile-verified
//
#include <hip/hip_runtime.h>
#include <math.h>

typedef __attribute__((ext_vector_type(2))) float v2f;
typedef __attribute__((ext_vector_type(8))) float v8f;

#define D_IN  64
#define NROW  512
#define KTOP  12
#define SCALEF 0.125f   // 1/sqrt(64)

__global__ __launch_bounds__(512)
void sparse_attn_topk(const float* __restrict__ x,
                      const float* __restrict__ flat,
                      float* __restrict__ out) {
  // LDS: K (pre-scaled) and Q rows in row-major [512][4] -> each lane's WMMA
  // operand is one aligned 8-byte load; plus weights and column accumulator.
  __shared__ __align__(16) float s_k[NROW][4];
  __shared__ __align__(16) float s_q[NROW][4];
  __shared__ __align__(16) float s_w[2 * D_IN * 4];   // wk then wq, row-major (c,d)
  __shared__ __align__(16) float s_col[NROW];

  const int tid = threadIdx.x;   // 0..511
  const int b   = blockIdx.x;    // 0..127  (N*T)

  // ---- phase 0: stage weights, zero column accumulator -------------------
  s_w[tid]   = flat[tid];
  s_col[tid] = 0.0f;
  __syncthreads();

  // ---- projections: one thread = one row of x ----------------------------
  {
    const float4* xr = (const float4*)(x + ((size_t)b * NROW + tid) * D_IN);
    float ka[4] = {0.f, 0.f, 0.f, 0.f};
    float qa[4] = {0.f, 0.f, 0.f, 0.f};
#pragma unroll
    for (int c = 0; c < D_IN; c += 4) {
      float4 xv = xr[c >> 2];
#pragma unroll
      for (int d = 0; d < 4; ++d) {
        ka[d] += xv.x * s_w[(c + 0) * 4 + d] + xv.y * s_w[(c + 1) * 4 + d]
               + xv.z * s_w[(c + 2) * 4 + d] + xv.w * s_w[(c + 3) * 4 + d];
        qa[d] += xv.x * s_w[D_IN * 4 + (c + 0) * 4 + d]
               + xv.y * s_w[D_IN * 4 + (c + 1) * 4 + d]
               + xv.z * s_w[D_IN * 4 + (c + 2) * 4 + d]
               + xv.w * s_w[D_IN * 4 + (c + 3) * 4 + d];
      }
    }
#pragma unroll
    for (int d = 0; d < 4; ++d) {
      s_k[tid][d] = SCALEF * ka[d];   // fold softmax scale into K
      s_q[tid][d] = qa[d];
    }
  }
  __syncthreads();

  // ---- per-wave WMMA sweeps ---------------------------------------------
  const int lane = tid & 31;
  const int wv   = tid >> 5;     // wave id 0..15
  const int half = lane >> 4;    // A/B operand half: elems {0,1} vs {2,3}
  const int r    = lane & 15;

  for (int rbi = 0; rbi < 2; ++rbi) {
    const int i0 = (wv + 16 * rbi) * 16;     // this wave's 16 rows
    v2f a = *(const v2f*)&s_k[i0 + r][2 * half];

    float m8[8], z8[8];
#pragma unroll
    for (int v = 0; v < 8; ++v) m8[v] = -__builtin_inff();

    // pass 1: row maxima (per-lane running max, cross-lane reduce at end)
    for (int j0 = 0; j0 < NROW; j0 += 16) {
      v2f bm = *(const v2f*)&s_q[j0 + r][2 * half];
      v8f c = {};
      c = __builtin_amdgcn_wmma_f32_16x16x4_f32(false, a, false, bm,
                                                (short)0, c, false, false);
#pragma unroll
      for (int v = 0; v < 8; ++v) m8[v] = fmaxf(m8[v], c[v]);
    }
#pragma unroll
    for (int v = 0; v < 8; ++v) {
      float mv = m8[v];
      mv = fmaxf(mv, __shfl_xor(mv, 1, 32));
      mv = fmaxf(mv, __shfl_xor(mv, 2, 32));
      mv = fmaxf(mv, __shfl_xor(mv, 4, 32));
      mv = fmaxf(mv, __shfl_xor(mv, 8, 32));   // reduce within 16-lane half
      m8[v] = mv;
      z8[v] = 0.0f;
    }

    // pass 2: row exp-sums
    for (int j0 = 0; j0 < NROW; j0 += 16) {
      v2f bm = *(const v2f*)&s_q[j0 + r][2 * half];
      v8f c = {};
      c = __builtin_amdgcn_wmma_f32_16x16x4_f32(false, a, false, bm,
                                                (short)0, c, false, false);
#pragma unroll
      for (int v = 0; v < 8; ++v) z8[v] += __expf(c[v] - m8[v]);
    }
#pragma unroll
    for (int v = 0; v < 8; ++v) {
      float zv = z8[v];
      zv += __shfl_xor(zv, 1, 32);
      zv += __shfl_xor(zv, 2, 32);
      zv += __shfl_xor(zv, 4, 32);
      zv += __shfl_xor(zv, 8, 32);
      z8[v] = __builtin_amdgcn_rcpf(zv);       // 1/Z per row
    }

    // pass 3: softmax column partials -> LDS atomic accumulation
    for (int j0 = 0; j0 < NROW; j0 += 16) {
      v2f bm = *(const v2f*)&s_q[j0 + r][2 * half];
      v8f c = {};
      c = __builtin_amdgcn_wmma_f32_16x16x4_f32(false, a, false, bm,
                                                (short)0, c, false, false);
      float acc = 0.0f;
#pragma unroll
      for (int v = 0; v < 8; ++v) acc += __expf(c[v] - m8[v]) * z8[v];
      atomicAdd(&s_col[j0 + r], acc);          // ds_add_f32
    }
  }
  __syncthreads();

  // ---- top-12 argmax rounds on one wave ---------------------------------
  if (tid < 32) {
    for (int t = 0; t < KTOP; ++t) {
      float bv = -__builtin_inff();
      int   bi = NROW;
      for (int j = tid; j < NROW; j += 32) {
        float v = s_col[j];
        if (v > bv) { bv = v; bi = j; }        // strict > keeps lowest index
      }
#pragma unroll
      for (int mask = 16; mask >= 1; mask >>= 1) {
        float ov = __shfl_xor(bv, mask, 32);
        int   oi = __shfl_xor(bi, mask, 32);
        if (ov > bv || (ov == bv && oi < bi)) { bv = ov; bi = oi; }
      }
      if (tid == 0) {
        out[(size_t)b * KTOP + t] = (float)bi;
        s_col[bi] = -__builtin_inff();         // knock out the winner
      }
      asm volatile("s_wait_dscnt 0x0" ::: "memory"); // order lane0 store vs next reads
    }
  }
}

extern "C" void kernel_launch(void* const* d_in, const int* in_sizes, int n_in,
                              void* d_out, int out_size, void* d_ws, size_t ws_size,
                              hipStream_t stream) {
  const float* x    = (const float*)d_in[0];
  const float* flat = (const float*)d_in[1];
  float* out        = (float*)d_out;
  const int nb = in_sizes[0] / (NROW * D_IN);  // N*T = 128 workgroups
  sparse_attn_topk<<<nb, 512, 0, stream>>>(x, flat, out);
}